// CRFLayer_14147622273590
// MI455X (gfx1250) — compile-verified
//
#include <hip/hip_runtime.h>

typedef float v2f __attribute__((ext_vector_type(2)));
typedef float v8f __attribute__((ext_vector_type(8)));

#define TT 32
#define START_TAG 30
#define STOP_TAG 31

// One wave (32 lanes) owns a tile of 16 batch rows.
// State W is LINEAR, rescaled by max(W) every step, running logZ:
//   W layout (WMMA C/D layout, lane L, hh=L>>4, bl=L&15):
//     reg j<8  : W[b=bl][to = 8*hh + j]
//     reg j>=8 : W[b=bl][to = 16 + 8*hh + (j-8)]
__global__ __launch_bounds__(128, 1) void crf_fwd_gold(
    const float* __restrict__ feats,
    const int*   __restrict__ tags,
    const int*   __restrict__ lengths,
    const float* __restrict__ trans,
    float*       __restrict__ res,   // [B] per-batch (forward - gold)
    int B, int S)
{
  __shared__ float sT[TT * TT];     // raw transitions (gold + terminal)
  __shared__ float sE[32 * 32];     // exp(trans) pre-packed in per-lane A-layout
  for (int i = threadIdx.x; i < TT * TT; i += blockDim.x) sT[i] = trans[i];
  if (threadIdx.x < 32) {
    const int l = threadIdx.x, lbl = l & 15, lhh = l >> 4;
#pragma unroll
    for (int k = 0; k < 8; ++k) {
      const int frm = 4 * k + 2 * lhh;
      sE[l * 32 + 2 * k + 0]      = __expf(trans[lbl * TT + frm]);
      sE[l * 32 + 2 * k + 1]      = __expf(trans[lbl * TT + frm + 1]);
      sE[l * 32 + 16 + 2 * k + 0] = __expf(trans[(lbl + 16) * TT + frm]);
      sE[l * 32 + 16 + 2 * k + 1] = __expf(trans[(lbl + 16) * TT + frm + 1]);
    }
  }
  __syncthreads();

  const int lane = threadIdx.x & 31;
  const int wid  = threadIdx.x >> 5;
  const int tile = blockIdx.x * 4 + wid;
  const int bl   = lane & 15;
  const int hh   = lane >> 4;
  const int b    = tile * 16 + bl;
  const float* sErow = &sE[lane * 32];

  const int len = lengths[b];
  int ml = len;
#pragma unroll
  for (int d = 1; d < 16; d <<= 1) {
    const int o = __shfl_xor(ml, d, 32);
    ml = (o > ml) ? o : ml;
  }
  const int maxLen = __builtin_amdgcn_readfirstlane(ml);

  const float* fb    = feats + (size_t)b * S * TT + 8 * hh;
  const float* fbrow = feats + (size_t)b * S * TT;
  const int*   tb    = tags  + (size_t)b * S;

  float W[16], fvP[16];
#pragma unroll
  for (int j = 0; j < 16; ++j) { W[j] = 0.0f; fvP[j] = 1.0f; }
  W[14] = hh ? 1.0f : 0.0f;   // to==START(30) lives in reg 14 of hh==1 lanes
  float logZ = 0.0f, fz = 0.0f;

  // prefetch step 0 emissions + tag
  float4 e0 = *(const float4*)(fb + 0);
  float4 e1 = *(const float4*)(fb + 4);
  float4 e2 = *(const float4*)(fb + 16);
  float4 e3 = *(const float4*)(fb + 20);
  int   tg   = tb[0];
  int   prev = START_TAG;
  float gold = 0.0f;

  for (int s = 0; s < maxLen; ++s) {
    // ---- prefetch next step (overlaps the whole compute chain) ----
    const int sn = (s + 1 < S) ? (s + 1) : s;
    const float* fn = fb + (size_t)sn * TT;
    float4 n0 = *(const float4*)(fn + 0);
    float4 n1 = *(const float4*)(fn + 4);
    float4 n2 = *(const float4*)(fn + 16);
    float4 n3 = *(const float4*)(fn + 20);
    const int   tgn   = tb[sn];
    const float gfeat = fbrow[(size_t)s * TT + tg];

    // ---- exp(emissions): off the recurrence chain ----
    float eE[16];
    eE[0]=__expf(e0.x);  eE[1]=__expf(e0.y);  eE[2]=__expf(e0.z);  eE[3]=__expf(e0.w);
    eE[4]=__expf(e1.x);  eE[5]=__expf(e1.y);  eE[6]=__expf(e1.z);  eE[7]=__expf(e1.w);
    eE[8]=__expf(e2.x);  eE[9]=__expf(e2.y);  eE[10]=__expf(e2.z); eE[11]=__expf(e2.w);
    eE[12]=__expf(e3.x); eE[13]=__expf(e3.y); eE[14]=__expf(e3.z); eE[15]=__expf(e3.w);

    // ---- normalization factors from CURRENT W: runs in the WMMA shadow ----
    float mx[8];
#pragma unroll
    for (int j = 0; j < 8; ++j) mx[j] = fmaxf(W[j], W[j + 8]);
#pragma unroll
    for (int j = 0; j < 4; ++j) mx[j] = fmaxf(mx[j], mx[j + 4]);
    mx[0] = fmaxf(mx[0], mx[2]);
    mx[1] = fmaxf(mx[1], mx[3]);
    float m = fmaxf(mx[0], mx[1]);
    m = fmaxf(m, __shfl_xor(m, 16, 32));
    const float inv   = __builtin_amdgcn_rcpf(m);
    const float logZn = logZ + __logf(m);
    float f[16];
#pragma unroll
    for (int j = 0; j < 16; ++j) f[j] = eE[j] * inv;   // fold rescale into emit

    // ---- E = exp(trans) x W via 16x WMMA f32 16x16x4 (4 chains, depth 4) ----
    v8f ct0, ct1, cb0, cb1;
#pragma unroll
    for (int i = 0; i < 8; ++i) { ct0[i]=0.0f; ct1[i]=0.0f; cb0[i]=0.0f; cb1[i]=0.0f; }

#pragma unroll
    for (int k = 0; k < 8; ++k) {
      // frm = 4k + 2*hh + v; the value lives in half-wave hsrc=(k>>1)&1.
      const int hsrc = (k >> 1) & 1;
      const int fo = 4 * k + 2 * hsrc;        // frm if this lane's half matches
      const int fr = 4 * k + 2 * (1 - hsrc);  // frm the other half needs
      const int io = (fo & 7) + 8 * ((fo >> 4) & 1);
      const int ir = (fr & 7) + 8 * ((fr >> 4) & 1);
      const float g0 = __shfl_xor(W[ir],     16, 32);
      const float g1 = __shfl_xor(W[ir + 1], 16, 32);
      v2f bk;
      bk.x = (hh == hsrc) ? W[io]     : g0;
      bk.y = (hh == hsrc) ? W[io + 1] : g1;
      const v2f atk = *(const v2f*)(sErow + 2 * k);
      const v2f abk = *(const v2f*)(sErow + 16 + 2 * k);
      if (k & 1) {
        ct1 = __builtin_amdgcn_wmma_f32_16x16x4_f32(false, atk, false, bk,
                                                    (short)0, ct1, false, false);
        cb1 = __builtin_amdgcn_wmma_f32_16x16x4_f32(false, abk, false, bk,
                                                    (short)0, cb1, false, false);
      } else {
        ct0 = __builtin_amdgcn_wmma_f32_16x16x4_f32(false, atk, false, bk,
                                                    (short)0, ct0, false, false);
        cb0 = __builtin_amdgcn_wmma_f32_16x16x4_f32(false, abk, false, bk,
                                                    (short)0, cb0, false, false);
      }
    }

    // ---- W' = (E_even + E_odd) * (exp(emit) * rcp(m)) ----
    float Wn[16];
#pragma unroll
    for (int j = 0; j < 16; ++j) {
      const float ev = (j < 8) ? (ct0[j] + ct1[j]) : (cb0[j - 8] + cb1[j - 8]);
      Wn[j] = ev * f[j];
    }

    // ---- capture linear snapshot at s == len-1 ----
    const int cap = (s == len - 1);
#pragma unroll
    for (int j = 0; j < 16; ++j) fvP[j] = cap ? Wn[j] : fvP[j];
    fz = cap ? logZn : fz;

    // ---- gold path score (branchless, identical in both half-waves) ----
    const int st = (s == 0) ? START_TAG : prev;
    float gadd = (s < len) ? (sT[tg * TT + st] + gfeat) : 0.0f;
    gadd += cap ? sT[STOP_TAG * TT + tg] : 0.0f;
    gold += gadd;

#pragma unroll
    for (int j = 0; j < 16; ++j) W[j] = Wn[j];
    logZ = logZn;
    prev = tg; tg = tgn;
    e0 = n0; e1 = n1; e2 = n2; e3 = n3;
  }

  // ---- terminal logsumexp(log(fvP) + fz + trans[STOP,:]) over 32 tags ----
  float term[16];
#pragma unroll
  for (int j = 0; j < 16; ++j) {
    const int to = (j < 8) ? (8 * hh + j) : (16 + 8 * hh + (j - 8));
    term[j] = __logf(fvP[j]) + fz + sT[STOP_TAG * TT + to];
  }
  float tm = term[0];
#pragma unroll
  for (int j = 1; j < 16; ++j) tm = fmaxf(tm, term[j]);
  tm = fmaxf(tm, __shfl_xor(tm, 16, 32));
  float ss = 0.0f;
#pragma unroll
  for (int j = 0; j < 16; ++j) ss += __expf(term[j] - tm);
  ss += __shfl_xor(ss, 16, 32);
  const float fwd = __logf(ss) + tm;

  if (lane < 16) res[b] = fwd - gold;
}

// Deterministic fixed-order tree reduction -> mean
__global__ __launch_bounds__(256) void crf_reduce_mean(
    const float* __restrict__ res, float* __restrict__ out, int B)
{
  __shared__ float sbuf[256];
  float s = 0.0f;
  for (int i = threadIdx.x; i < B; i += 256) s += res[i];
  sbuf[threadIdx.x] = s;
  __syncthreads();
  for (int off = 128; off > 0; off >>= 1) {
    if (threadIdx.x < off) sbuf[threadIdx.x] += sbuf[threadIdx.x + off];
    __syncthreads();
  }
  if (threadIdx.x == 0) out[0] = sbuf[0] / (float)B;
}

extern "C" void kernel_launch(void* const* d_in, const int* in_sizes, int n_in,
                              void* d_out, int out_size, void* d_ws, size_t ws_size,
                              hipStream_t stream) {
  const float* feats   = (const float*)d_in[0];
  const int*   tags    = (const int*)  d_in[1];
  const int*   lengths = (const int*)  d_in[2];
  const float* trans   = (const float*)d_in[3];
  const int B = in_sizes[2];            // lengths: [B]
  const int S = in_sizes[1] / B;        // tags: [B, S]
  float* res = (float*)d_ws;            // B floats of scratch

  const int tiles  = B / 16;            // 16 batches per wave
  const int blocks = tiles / 4;         // 4 waves per block (blockDim 128)
  crf_fwd_gold<<<blocks, 128, 0, stream>>>(feats, tags, lengths, trans, res, B, S);
  crf_reduce_mean<<<1, 256, 0, stream>>>(res, (float*)d_out, B);
}